// MulticlassPrototypicalClassifier_68049461838412
// MI455X (gfx1250) — compile-verified
//
#include <hip/hip_runtime.h>
#include <math.h>

typedef __attribute__((ext_vector_type(2))) float v2f;
typedef __attribute__((ext_vector_type(8))) float v8f;

#define N_TOT 1536
#define D_IN  256
#define D_OUT 64
#define N_WAY 16
#define N_SUP 768
#define SLOPE 0.2f

// ---------------------------------------------------------------------------
// Kernel 1: g = hidden @ W^T   (M=1536, K=256, N=64) via V_WMMA_F32_16X16X4_F32
// One wave per 16x16 output tile; 64 WMMA issues per wave (K=256 / 4).
// Both A and B fragments are contiguous 8-byte loads (hidden and W row-major).
// ---------------------------------------------------------------------------
__global__ void __launch_bounds__(32)
gemm_g_wmma(const float* __restrict__ hidden, const float* __restrict__ W,
            float* __restrict__ g) {
  const int lane = threadIdx.x & 31;
  const int l  = lane & 15;
  const int hi = lane >> 4;                 // 0: K=k0,k0+1   1: K=k0+2,k0+3
  const int rowbase = blockIdx.x * 16;
  const int colbase = blockIdx.y * 16;
  const float* arow = hidden + (size_t)(rowbase + l) * D_IN + 2 * hi;
  const float* brow = W      + (size_t)(colbase + l) * D_IN + 2 * hi;   // B[k][n] = W[n][k]
  v8f acc = {0.f, 0.f, 0.f, 0.f, 0.f, 0.f, 0.f, 0.f};
#pragma unroll 8
  for (int k0 = 0; k0 < D_IN; k0 += 4) {
    v2f av = *(const v2f*)(arow + k0);
    v2f bv = *(const v2f*)(brow + k0);
    acc = __builtin_amdgcn_wmma_f32_16x16x4_f32(false, av, false, bv,
                                                (short)0, acc, false, false);
  }
#pragma unroll
  for (int r = 0; r < 8; ++r)
    g[(size_t)(rowbase + r + 8 * hi) * D_OUT + colbase + l] = acc[r];
}

// ---------------------------------------------------------------------------
// Kernel 2: attention logits with fused mask, written into the 'a' region.
//   e[i,j] = sum_d lrelu(g[i,d] + g[j,d]) * attn_w[d]
//   e[i,j] = -inf  if known[i] && known[j] && labels[i] != labels[j]
// 16 i-rows per block; j tiled by 16 through LDS. Stride-68 LDS rows keep the
// 16 gI rows on distinct banks (68 mod 64 = 4) and 16B-aligned for float4.
// ---------------------------------------------------------------------------
__global__ void __launch_bounds__(256)
attn_logits(const float* __restrict__ g, const float* __restrict__ attn_w,
            const int* __restrict__ labels, const unsigned char* __restrict__ zda,
            float* __restrict__ e_out) {
  __shared__ __align__(16) float gI[16 * 68];
  __shared__ __align__(16) float gJ[16 * 68];
  __shared__ __align__(16) float ws[64];
  const int tid = threadIdx.x;
  const int ti = tid & 15, tj = tid >> 4;
  const int ibase = blockIdx.x * 16;

  for (int idx = tid; idx < 16 * 64; idx += 256)
    gI[(idx >> 6) * 68 + (idx & 63)] = g[(size_t)(ibase + (idx >> 6)) * D_OUT + (idx & 63)];
  if (tid < 64) ws[tid] = attn_w[tid];

  const int i     = ibase + ti;
  const int labi  = labels[i];
  const bool ki   = (zda[i] == 0);
  __syncthreads();

  for (int jt = 0; jt < N_TOT / 16; ++jt) {
    for (int idx = tid; idx < 16 * 64; idx += 256)
      gJ[(idx >> 6) * 68 + (idx & 63)] = g[(size_t)(jt * 16 + (idx >> 6)) * D_OUT + (idx & 63)];
    __syncthreads();

    float s = 0.f;
#pragma unroll
    for (int d = 0; d < 64; d += 4) {
      float4 xi = *(const float4*)&gI[ti * 68 + d];
      float4 xj = *(const float4*)&gJ[tj * 68 + d];
      float4 w4 = *(const float4*)&ws[d];
      float x0 = xi.x + xj.x; s = fmaf(x0 >= 0.f ? x0 : SLOPE * x0, w4.x, s);
      float x1 = xi.y + xj.y; s = fmaf(x1 >= 0.f ? x1 : SLOPE * x1, w4.y, s);
      float x2 = xi.z + xj.z; s = fmaf(x2 >= 0.f ? x2 : SLOPE * x2, w4.z, s);
      float x3 = xi.w + xj.w; s = fmaf(x3 >= 0.f ? x3 : SLOPE * x3, w4.w, s);
    }
    const int j = jt * 16 + tj;
    const bool kj = (zda[j] == 0);
    if (ki && kj && (labi != labels[j])) s = -__builtin_inff();
    e_out[(size_t)i * N_TOT + j] = s;
    __syncthreads();
  }
}

// ---------------------------------------------------------------------------
// Kernel 3: in-place row softmax over 'a' (one block per row).
// Rows always contain a finite diagonal entry, so no all -inf rows.
// ---------------------------------------------------------------------------
__global__ void __launch_bounds__(256)
softmax_rows(float* __restrict__ a) {
  __shared__ float red[256];
  float* r = a + (size_t)blockIdx.x * N_TOT;
  const int tid = threadIdx.x;

  float m = -__builtin_inff();
  for (int j = tid; j < N_TOT; j += 256) m = fmaxf(m, r[j]);
  red[tid] = m; __syncthreads();
  for (int s = 128; s > 0; s >>= 1) {
    if (tid < s) red[tid] = fmaxf(red[tid], red[tid + s]);
    __syncthreads();
  }
  m = red[0]; __syncthreads();

  float sum = 0.f;
  for (int j = tid; j < N_TOT; j += 256) {
    float t = __expf(r[j] - m);
    r[j] = t;
    sum += t;
  }
  red[tid] = sum; __syncthreads();
  for (int s = 128; s > 0; s >>= 1) {
    if (tid < s) red[tid] += red[tid + s];
    __syncthreads();
  }
  const float inv = 1.f / red[0];
  for (int j = tid; j < N_TOT; j += 256) r[j] *= inv;
}

// ---------------------------------------------------------------------------
// Kernel 4: h_out = a @ g   (M=1536, K=1536, N=64) via V_WMMA_F32_16X16X4_F32.
// 384 WMMA issues per wave. 'a' (9.4 MB) is L2-resident on MI455X (192 MB L2).
// ---------------------------------------------------------------------------
__global__ void __launch_bounds__(32)
gemm_hout_wmma(const float* __restrict__ a, const float* __restrict__ g,
               float* __restrict__ h_out) {
  const int lane = threadIdx.x & 31;
  const int l  = lane & 15;
  const int hi = lane >> 4;
  const int rowbase = blockIdx.x * 16;
  const int colbase = blockIdx.y * 16;
  const float* arow = a + (size_t)(rowbase + l) * N_TOT + 2 * hi;
  v8f acc = {0.f, 0.f, 0.f, 0.f, 0.f, 0.f, 0.f, 0.f};
#pragma unroll 4
  for (int k0 = 0; k0 < N_TOT; k0 += 4) {
    v2f av = *(const v2f*)(arow + k0);
    v2f bv;                                            // B = g (row-major KxN)
    bv.x = g[(size_t)(k0 + 2 * hi)     * D_OUT + colbase + l];
    bv.y = g[(size_t)(k0 + 2 * hi + 1) * D_OUT + colbase + l];
    acc = __builtin_amdgcn_wmma_f32_16x16x4_f32(false, av, false, bv,
                                                (short)0, acc, false, false);
  }
#pragma unroll
  for (int r = 0; r < 8; ++r)
    h_out[(size_t)(rowbase + r + 8 * hi) * D_OUT + colbase + l] = acc[r];
}

// ---------------------------------------------------------------------------
// Kernel 5: per-class centroids over the support half (all labels, per ref).
// ---------------------------------------------------------------------------
__global__ void __launch_bounds__(64)
centroid_kernel(const float* __restrict__ h_out, const int* __restrict__ labels,
                float* __restrict__ centroids) {
  const int c = blockIdx.x;
  const int d = threadIdx.x;
  float sum = 0.f;
  int cnt = 0;
  for (int i = 0; i < N_SUP; ++i) {
    if (labels[i] == c) { sum += h_out[(size_t)i * D_OUT + d]; ++cnt; }
  }
  centroids[c * D_OUT + d] = (cnt > 0) ? (sum / (float)cnt) : 0.f;
}

// ---------------------------------------------------------------------------
// Kernel 6: scores[q,c] = 1 / (||h_q - centroid_c|| + 1e-10)
// ---------------------------------------------------------------------------
__global__ void __launch_bounds__(256)
scores_kernel(const float* __restrict__ h_out, const float* __restrict__ centroids,
              float* __restrict__ scores) {
  const int idx = blockIdx.x * 256 + threadIdx.x;   // 768*16 threads
  const int q = idx >> 4, c = idx & 15;
  const float* hq = h_out + (size_t)(N_SUP + q) * D_OUT;
  const float* ce = centroids + c * D_OUT;
  float hh = 0.f, cc2 = 0.f, hc = 0.f;
#pragma unroll 8
  for (int d = 0; d < D_OUT; ++d) {
    const float hv = hq[d], cv = ce[d];
    hh  = fmaf(hv, hv, hh);
    cc2 = fmaf(cv, cv, cc2);
    hc  = fmaf(hv, cv, hc);
  }
  const float d2   = hh + cc2 - 2.f * hc;
  const float dist = sqrtf(fmaxf(d2, 0.f));
  scores[idx] = 1.f / (dist + 1e-10f);
}

// ---------------------------------------------------------------------------
extern "C" void kernel_launch(void* const* d_in, const int* in_sizes, int n_in,
                              void* d_out, int out_size, void* d_ws, size_t ws_size,
                              hipStream_t stream) {
  const float*         hidden = (const float*)d_in[0];          // 1536x256
  const float*         W      = (const float*)d_in[1];          // 64x256
  const float*         attn_w = (const float*)d_in[2];          // 64
  const int*           labels = (const int*)d_in[3];            // 1536
  const unsigned char* zda    = (const unsigned char*)d_in[4];  // 1536 (bool)
  // d_in[5] query_mask and d_in[6] known_attacks_count unused (fixed split / N_WAY)

  float* out    = (float*)d_out;
  float* scores = out;                               // 768*16
  float* h_out  = out + N_SUP * N_WAY;               // 1536*64
  float* a      = h_out + N_TOT * D_OUT;             // 1536*1536

  float* g         = (float*)d_ws;                   // 1536*64 floats
  float* centroids = g + N_TOT * D_OUT;              // 16*64 floats

  gemm_g_wmma   <<<dim3(N_TOT / 16, D_OUT / 16), 32,  0, stream>>>(hidden, W, g);
  attn_logits   <<<N_TOT / 16,                    256, 0, stream>>>(g, attn_w, labels, zda, a);
  softmax_rows  <<<N_TOT,                         256, 0, stream>>>(a);
  gemm_hout_wmma<<<dim3(N_TOT / 16, D_OUT / 16), 32,  0, stream>>>(a, g, h_out);
  centroid_kernel<<<N_WAY, D_OUT, 0, stream>>>(h_out, labels, centroids);
  scores_kernel <<<(N_SUP * N_WAY) / 256, 256, 0, stream>>>(h_out, centroids, scores);
}